// YoshidaIntegrator_72593537237646
// MI455X (gfx1250) — compile-verified
//
#include <hip/hip_runtime.h>
#include <cmath>

typedef float v2f __attribute__((ext_vector_type(2)));
typedef float v8f __attribute__((ext_vector_type(8)));

#define B_DIM 8192
#define D_DIM 1024
#define R_DIM 64

// ---------------------------------------------------------------------------
// Pack a row-major [K, N] B-matrix into WMMA-fragment order so that one
// global_load_b128 per lane covers TWO k-chunks (K=8) of B-fragment data.
//   col  = nt*16 + (lane&15),  h = (lane>>4)*2
//   e0 = src[(8p+h  )*ldb + col]    e1 = src[(8p+h+1)*ldb + col]
//   e2 = src[(8p+4+h)*ldb + col]    e3 = src[(8p+5+h)*ldb + col]
// ---------------------------------------------------------------------------
__global__ void pack_b(float4* __restrict__ dst, const float* __restrict__ src,
                       int ldb, int nkpairs, int total) {
  int t = blockIdx.x * blockDim.x + threadIdx.x;
  if (t >= total) return;
  int lane = t & 31;
  int p    = (t >> 5) % nkpairs;
  int nt   = (t >> 5) / nkpairs;
  int col  = nt * 16 + (lane & 15);
  int h    = (lane >> 4) << 1;
  int r0   = 8 * p + h;
  float4 o;
  o.x = src[(size_t)(r0)     * ldb + col];
  o.y = src[(size_t)(r0 + 1) * ldb + col];
  o.z = src[(size_t)(r0 + 4) * ldb + col];
  o.w = src[(size_t)(r0 + 5) * ldb + col];
  dst[t] = o;
}

// ---------------------------------------------------------------------------
// Elementwise drift: x += cdt * v   (in-place safe; float4 vectorized)
// ---------------------------------------------------------------------------
__global__ void yoshida_xupd(float* __restrict__ xout,
                             const float* __restrict__ xin,
                             const float* __restrict__ v,
                             float cdt, int n4) {
  int i = blockIdx.x * blockDim.x + threadIdx.x;
  if (i < n4) {
    float4 a = ((const float4*)xin)[i];
    float4 b = ((const float4*)v)[i];
    a.x += cdt * b.x; a.y += cdt * b.y; a.z += cdt * b.z; a.w += cdt * b.w;
    ((float4*)xout)[i] = a;
  }
}

// ---------------------------------------------------------------------------
// q' = (v@U)^2 * (1 + 0.1*tanh(x@V_w)) * sing(x)       [B, R]
// One wave computes a 16x64 tile via V_WMMA_F32_16X16X4_F32.
// 2 waves/block (64 thr) -> 32 rows/block, grid = 256 blocks (WGP spread).
// Per k-pair: all 12 fragment loads issued first (named locals), then 16 WMMAs.
// ---------------------------------------------------------------------------
__global__ void yoshida_q(float* __restrict__ q,
                          const float* __restrict__ x,
                          const float* __restrict__ v,
                          const float4* __restrict__ PU,
                          const float4* __restrict__ PVw) {
  const int lane    = threadIdx.x & 31;
  const int wave    = threadIdx.x >> 5;
  const int rowbase = blockIdx.x * 32 + wave * 16;
  const int mrow    = lane & 15;          // A-fragment row
  const int koff    = (lane >> 4) << 1;   // 0 (lanes 0-15) or 2 (lanes 16-31)

  const float* vrow = v + (size_t)(rowbase + mrow) * D_DIM + koff;
  const float* xrow = x + (size_t)(rowbase + mrow) * D_DIM + koff;

  v8f P0 = {}, P1 = {}, P2 = {}, P3 = {};   // accum for v@U
  v8f T0 = {}, T1 = {}, T2 = {}, T3 = {};   // accum for x@V_w
  float xsq = 0.0f;

  const int NKP = D_DIM / 8;               // 128 k-pairs
  const float4* pu = PU  + lane;
  const float4* pv = PVw + lane;

#pragma unroll 2
  for (int p = 0; p < NKP; ++p) {
    const int k = p * 8;
    // ---- issue ALL loads for this k-pair (distinct locals) ----
    v2f av0 = *(const v2f*)(vrow + k);
    v2f av1 = *(const v2f*)(vrow + k + 4);
    v2f ax0 = *(const v2f*)(xrow + k);
    v2f ax1 = *(const v2f*)(xrow + k + 4);
    float4 bu0 = pu[(size_t)(0 * NKP + p) * 32];
    float4 bu1 = pu[(size_t)(1 * NKP + p) * 32];
    float4 bu2 = pu[(size_t)(2 * NKP + p) * 32];
    float4 bu3 = pu[(size_t)(3 * NKP + p) * 32];
    float4 bv0 = pv[(size_t)(0 * NKP + p) * 32];
    float4 bv1 = pv[(size_t)(1 * NKP + p) * 32];
    float4 bv2 = pv[(size_t)(2 * NKP + p) * 32];
    float4 bv3 = pv[(size_t)(3 * NKP + p) * 32];

    xsq += ax0.x * ax0.x + ax0.y * ax0.y + ax1.x * ax1.x + ax1.y * ax1.y;

    // ---- 16 WMMAs ----
    P0 = __builtin_amdgcn_wmma_f32_16x16x4_f32(false, av0, false, (v2f){bu0.x, bu0.y}, (short)0, P0, false, false);
    P0 = __builtin_amdgcn_wmma_f32_16x16x4_f32(false, av1, false, (v2f){bu0.z, bu0.w}, (short)0, P0, false, false);
    P1 = __builtin_amdgcn_wmma_f32_16x16x4_f32(false, av0, false, (v2f){bu1.x, bu1.y}, (short)0, P1, false, false);
    P1 = __builtin_amdgcn_wmma_f32_16x16x4_f32(false, av1, false, (v2f){bu1.z, bu1.w}, (short)0, P1, false, false);
    P2 = __builtin_amdgcn_wmma_f32_16x16x4_f32(false, av0, false, (v2f){bu2.x, bu2.y}, (short)0, P2, false, false);
    P2 = __builtin_amdgcn_wmma_f32_16x16x4_f32(false, av1, false, (v2f){bu2.z, bu2.w}, (short)0, P2, false, false);
    P3 = __builtin_amdgcn_wmma_f32_16x16x4_f32(false, av0, false, (v2f){bu3.x, bu3.y}, (short)0, P3, false, false);
    P3 = __builtin_amdgcn_wmma_f32_16x16x4_f32(false, av1, false, (v2f){bu3.z, bu3.w}, (short)0, P3, false, false);
    T0 = __builtin_amdgcn_wmma_f32_16x16x4_f32(false, ax0, false, (v2f){bv0.x, bv0.y}, (short)0, T0, false, false);
    T0 = __builtin_amdgcn_wmma_f32_16x16x4_f32(false, ax1, false, (v2f){bv0.z, bv0.w}, (short)0, T0, false, false);
    T1 = __builtin_amdgcn_wmma_f32_16x16x4_f32(false, ax0, false, (v2f){bv1.x, bv1.y}, (short)0, T1, false, false);
    T1 = __builtin_amdgcn_wmma_f32_16x16x4_f32(false, ax1, false, (v2f){bv1.z, bv1.w}, (short)0, T1, false, false);
    T2 = __builtin_amdgcn_wmma_f32_16x16x4_f32(false, ax0, false, (v2f){bv2.x, bv2.y}, (short)0, T2, false, false);
    T2 = __builtin_amdgcn_wmma_f32_16x16x4_f32(false, ax1, false, (v2f){bv2.z, bv2.w}, (short)0, T2, false, false);
    T3 = __builtin_amdgcn_wmma_f32_16x16x4_f32(false, ax0, false, (v2f){bv3.x, bv3.y}, (short)0, T3, false, false);
    T3 = __builtin_amdgcn_wmma_f32_16x16x4_f32(false, ax1, false, (v2f){bv3.z, bv3.w}, (short)0, T3, false, false);
  }

  // full row sum of x^2: combine the two lane-halves (wave32)
  float xfull = xsq + __shfl_xor(xsq, 16, 32);
  // sing = 1 + 20*exp(-||x||^2/10); lane l holds value for row (l%16)
  float sing = 1.0f + 20.0f * expf(-xfull * 0.1f);

  v8f P[4] = {P0, P1, P2, P3};
  v8f T[4] = {T0, T1, T2, T3};
#pragma unroll
  for (int nt = 0; nt < 4; ++nt) {
#pragma unroll
    for (int g = 0; g < 8; ++g) {
      int m = g + ((lane >> 4) << 3);           // C/D row: g or g+8
      float s  = __shfl(sing, m, 32);           // row m's sing factor
      float pe = P[nt][g];
      float te = T[nt][g];
      float qe = pe * pe * (1.0f + 0.1f * tanhf(te)) * s;
      q[(size_t)(rowbase + m) * R_DIM + nt * 16 + mrow] = qe;
    }
  }
}

// ---------------------------------------------------------------------------
// v_new = v + ddt*(force - q'@W); optionally x += c4dt*v_new (final substep)
// Block of 256 threads: all 8 waves share 16 rows, each wave owns 128 cols.
// grid = B/16 = 512 blocks. K = 64 -> 8 packed k-pairs.
// ---------------------------------------------------------------------------
__global__ void yoshida_v(float* __restrict__ vout,
                          const float* __restrict__ vin,
                          const float* __restrict__ q,
                          const float4* __restrict__ PW,
                          const float* __restrict__ force,
                          float ddt, float c4dt,
                          float* __restrict__ x) {
  const int lane    = threadIdx.x & 31;
  const int wave    = threadIdx.x >> 5;
  const int rowbase = blockIdx.x * 16;
  const int colbase = wave * 128;
  const int mrow    = lane & 15;
  const int koff    = (lane >> 4) << 1;

  const float* qrow = q + (size_t)(rowbase + mrow) * R_DIM + koff;

  v8f acc[8] = {v8f{}, v8f{}, v8f{}, v8f{}, v8f{}, v8f{}, v8f{}, v8f{}};

  const int NKP = R_DIM / 8;               // 8 k-pairs
  const float4* pw = PW + (size_t)(wave * 8) * NKP * 32 + lane;

#pragma unroll 2
  for (int p = 0; p < NKP; ++p) {
    const int k = p * 8;
    v2f a0 = *(const v2f*)(qrow + k);
    v2f a1 = *(const v2f*)(qrow + k + 4);
    float4 bw0 = pw[(size_t)(0 * NKP + p) * 32];
    float4 bw1 = pw[(size_t)(1 * NKP + p) * 32];
    float4 bw2 = pw[(size_t)(2 * NKP + p) * 32];
    float4 bw3 = pw[(size_t)(3 * NKP + p) * 32];
    float4 bw4 = pw[(size_t)(4 * NKP + p) * 32];
    float4 bw5 = pw[(size_t)(5 * NKP + p) * 32];
    float4 bw6 = pw[(size_t)(6 * NKP + p) * 32];
    float4 bw7 = pw[(size_t)(7 * NKP + p) * 32];

    acc[0] = __builtin_amdgcn_wmma_f32_16x16x4_f32(false, a0, false, (v2f){bw0.x, bw0.y}, (short)0, acc[0], false, false);
    acc[0] = __builtin_amdgcn_wmma_f32_16x16x4_f32(false, a1, false, (v2f){bw0.z, bw0.w}, (short)0, acc[0], false, false);
    acc[1] = __builtin_amdgcn_wmma_f32_16x16x4_f32(false, a0, false, (v2f){bw1.x, bw1.y}, (short)0, acc[1], false, false);
    acc[1] = __builtin_amdgcn_wmma_f32_16x16x4_f32(false, a1, false, (v2f){bw1.z, bw1.w}, (short)0, acc[1], false, false);
    acc[2] = __builtin_amdgcn_wmma_f32_16x16x4_f32(false, a0, false, (v2f){bw2.x, bw2.y}, (short)0, acc[2], false, false);
    acc[2] = __builtin_amdgcn_wmma_f32_16x16x4_f32(false, a1, false, (v2f){bw2.z, bw2.w}, (short)0, acc[2], false, false);
    acc[3] = __builtin_amdgcn_wmma_f32_16x16x4_f32(false, a0, false, (v2f){bw3.x, bw3.y}, (short)0, acc[3], false, false);
    acc[3] = __builtin_amdgcn_wmma_f32_16x16x4_f32(false, a1, false, (v2f){bw3.z, bw3.w}, (short)0, acc[3], false, false);
    acc[4] = __builtin_amdgcn_wmma_f32_16x16x4_f32(false, a0, false, (v2f){bw4.x, bw4.y}, (short)0, acc[4], false, false);
    acc[4] = __builtin_amdgcn_wmma_f32_16x16x4_f32(false, a1, false, (v2f){bw4.z, bw4.w}, (short)0, acc[4], false, false);
    acc[5] = __builtin_amdgcn_wmma_f32_16x16x4_f32(false, a0, false, (v2f){bw5.x, bw5.y}, (short)0, acc[5], false, false);
    acc[5] = __builtin_amdgcn_wmma_f32_16x16x4_f32(false, a1, false, (v2f){bw5.z, bw5.w}, (short)0, acc[5], false, false);
    acc[6] = __builtin_amdgcn_wmma_f32_16x16x4_f32(false, a0, false, (v2f){bw6.x, bw6.y}, (short)0, acc[6], false, false);
    acc[6] = __builtin_amdgcn_wmma_f32_16x16x4_f32(false, a1, false, (v2f){bw6.z, bw6.w}, (short)0, acc[6], false, false);
    acc[7] = __builtin_amdgcn_wmma_f32_16x16x4_f32(false, a0, false, (v2f){bw7.x, bw7.y}, (short)0, acc[7], false, false);
    acc[7] = __builtin_amdgcn_wmma_f32_16x16x4_f32(false, a1, false, (v2f){bw7.z, bw7.w}, (short)0, acc[7], false, false);
  }

#pragma unroll
  for (int nt = 0; nt < 8; ++nt) {
#pragma unroll
    for (int g = 0; g < 8; ++g) {
      int m = g + ((lane >> 4) << 3);
      size_t idx = (size_t)(rowbase + m) * D_DIM + colbase + nt * 16 + mrow;
      float vn = vin[idx] + ddt * (force[idx] - acc[nt][g]);
      vout[idx] = vn;
      if (c4dt != 0.0f) x[idx] += c4dt * vn;   // final C4 drift fold
    }
  }
  (void)koff;
}

// ---------------------------------------------------------------------------
// Host orchestration: pack factors once, then 4 Yoshida steps x 3 substeps.
// d_ws layout: q [B*R] (2 MB) | PU (256 KB) | PVw (256 KB) | PW (256 KB)
// ---------------------------------------------------------------------------
extern "C" void kernel_launch(void* const* d_in, const int* in_sizes, int n_in,
                              void* d_out, int out_size, void* d_ws, size_t ws_size,
                              hipStream_t stream) {
  const float* x_in  = (const float*)d_in[0];
  const float* v_in  = (const float*)d_in[1];
  const float* force = (const float*)d_in[2];
  const float* U     = (const float*)d_in[3];
  const float* W     = (const float*)d_in[4];
  const float* Vw    = (const float*)d_in[5];
  (void)in_sizes; (void)n_in; (void)out_size; (void)ws_size;

  float* xo = (float*)d_out;                       // x state / final x
  float* vo = xo + (size_t)B_DIM * D_DIM;          // v state / final v

  char* ws = (char*)d_ws;
  float*  q   = (float*)ws;                                   // 2 MB
  float4* PU  = (float4*)(ws + (size_t)B_DIM * R_DIM * 4);    // 256 KB
  float4* PVw = PU  + (size_t)(D_DIM / 8) * 4 * 32;           // 256 KB
  float4* PW  = PVw + (size_t)(D_DIM / 8) * 4 * 32;           // 256 KB

  // One-time fragment packing (deterministic, graph-safe).
  {
    int totUV = (R_DIM / 16) * (D_DIM / 8) * 32;   // 4 tiles * 128 pairs * 32
    int totW  = (D_DIM / 16) * (R_DIM / 8) * 32;   // 64 tiles * 8 pairs * 32
    pack_b<<<(totUV + 255) / 256, 256, 0, stream>>>(PU,  U,  R_DIM, D_DIM / 8, totUV);
    pack_b<<<(totUV + 255) / 256, 256, 0, stream>>>(PVw, Vw, R_DIM, D_DIM / 8, totUV);
    pack_b<<<(totW  + 255) / 256, 256, 0, stream>>>(PW,  W,  D_DIM, R_DIM / 8, totW);
  }

  const double w1 = 1.0 / (2.0 - cbrt(2.0));
  const double w0 = -cbrt(2.0) * w1;
  const double dt = 0.01;
  const float Cdt[4] = {(float)(0.5 * w1 * dt), (float)(0.5 * (w0 + w1) * dt),
                        (float)(0.5 * (w0 + w1) * dt), (float)(0.5 * w1 * dt)};
  const float Ddt[3] = {(float)(w1 * dt), (float)(w0 * dt), (float)(w1 * dt)};

  const int steps = 4;                              // setup_inputs fixes steps=4
  const int n4 = B_DIM * D_DIM / 4;
  const int gx = (n4 + 255) / 256;

  bool first = true;
  for (int st = 0; st < steps; ++st) {
    for (int sub = 0; sub < 3; ++sub) {
      const bool last = (st == steps - 1) && (sub == 2);
      const float* xsrc = first ? x_in : xo;
      const float* vsrc = first ? v_in : vo;

      // drift: x += C*dt*v
      yoshida_xupd<<<gx, 256, 0, stream>>>(xo, xsrc, vsrc, Cdt[sub], n4);
      // q' = (v@U)^2 * plasticity(x@Vw) * sing(x)
      yoshida_q<<<B_DIM / 32, 64, 0, stream>>>(q, xo, vsrc, PU, PVw);
      // kick: v += D*dt*(force - q'@W); final substep also folds x += C4*dt*v
      yoshida_v<<<B_DIM / 16, 256, 0, stream>>>(vo, vsrc, q, PW, force,
                                                Ddt[sub], last ? Cdt[3] : 0.0f, xo);
      first = false;
    }
    if (st != steps - 1) {
      // trailing drift x += C4*dt*v between steps
      yoshida_xupd<<<gx, 256, 0, stream>>>(xo, xo, vo, Cdt[3], n4);
    }
  }
}